// YoloNMS_58171037057514
// MI455X (gfx1250) — compile-verified
//
#include <hip/hip_runtime.h>
#include <stdint.h>

#define NN 8192
#define NUM_CLASSES 80
#define ROWLEN 85
#define IOU_THR 0.5f

typedef float v2f __attribute__((ext_vector_type(2)));
typedef float v8f __attribute__((ext_vector_type(8)));

// ---- workspace layout (bytes) ----
#define OFF_MAXC   0
#define OFF_SCORES 256
#define OFF_CAT    (OFF_SCORES + NN*4)
#define OFF_ORDER  (OFF_CAT + NN*4)
#define OFF_AREA   (OFF_ORDER + NN*4)
#define OFF_BSORT  (OFF_AREA + NN*4)          // 16B aligned (131328 % 16 == 0)
#define OFF_KEEP   (OFF_BSORT + NN*16)
#define OFF_MASK   (OFF_KEEP + NN*4)          // uint32 mask[NN][256], 8 MB

// ---------------------------------------------------------------------------
__global__ void k_init(int* maxc_i) { *maxc_i = 0; }  // +0.0f as int bits

// per-row class max/argmax, combined score, global coord max (positive floats
// compare identically as ints -> integer atomicMax is exact)
__global__ void k_scores(const float* __restrict__ X, float* __restrict__ scores,
                         int* __restrict__ cat, int* __restrict__ maxc_i) {
  int r = blockIdx.x * blockDim.x + threadIdx.x;
  if (r >= NN) return;
  const float* row = X + (size_t)r * ROWLEN;
  float best = row[5]; int bi = 0;
  #pragma unroll 4
  for (int c = 1; c < NUM_CLASSES; ++c) {
    float v = row[5 + c];
    if (v > best) { best = v; bi = c; }   // first-occurrence argmax like jnp
  }
  scores[r] = row[4] * best;
  cat[r] = bi;
  float m = fmaxf(fmaxf(row[0], row[1]), fmaxf(row[2], row[3]));
  atomicMax(maxc_i, __float_as_int(m));
}

// single-workgroup bitonic sort of 8192 (score, idx) pairs, descending score.
// 64KB dynamic LDS (WGP has 320KB).
__global__ void k_sort(const float* __restrict__ scores, int* __restrict__ order) {
  extern __shared__ char smem[];
  float* key = (float*)smem;
  int*   val = (int*)(smem + NN * 4);
  const int tid = threadIdx.x;
  for (int i = tid; i < NN; i += 1024) { key[i] = scores[i]; val[i] = i; }
  __syncthreads();
  for (int k = 2; k <= NN; k <<= 1) {
    for (int j = k >> 1; j > 0; j >>= 1) {
      for (int i = tid; i < NN; i += 1024) {
        int l = i ^ j;
        if (l > i) {
          bool down = ((i & k) == 0);          // descending blocks -> final descending
          float ki = key[i], kl = key[l];
          bool sw = down ? (ki < kl) : (ki > kl);
          if (sw) {
            key[i] = kl; key[l] = ki;
            int t = val[i]; val[i] = val[l]; val[l] = t;
          }
        }
      }
      __syncthreads();
    }
  }
  for (int i = tid; i < NN; i += 1024) order[i] = val[i];
}

// gather sorted, class-offset boxes + areas
__global__ void k_gather(const float* __restrict__ X, const int* __restrict__ order,
                         const int* __restrict__ cat, const float* __restrict__ maxc,
                         float4* __restrict__ bsort, float* __restrict__ area) {
  int i = blockIdx.x * blockDim.x + threadIdx.x;
  if (i >= NN) return;
  int o = order[i];
  const float* row = X + (size_t)o * ROWLEN;
  float off = (float)cat[o] * (maxc[0] + 1.0f);
  float x0 = row[0] + off, y0 = row[1] + off;
  float x1 = row[2] + off, y1 = row[3] + off;
  bsort[i] = make_float4(x0, y0, x1, y1);
  area[i] = (x1 - x0) * (y1 - y0);
}

// area_i + area_j outer sum for a 16x16 tile via one V_WMMA_F32_16X16X4_F32,
// delivered directly in the 16x16 f32 C/D lane striping the IoU math uses.
__device__ __forceinline__ v8f area_outer_sum(int lane, float ai, float aj) {
#if __has_builtin(__builtin_amdgcn_wmma_f32_16x16x4_f32)
  // A (16x4): lanes 0-15 hold row M=lane: K0=area_i[M], K1=1; lanes 16-31: K2=K3=0
  v2f A; A[0] = (lane < 16) ? ai : 0.0f; A[1] = (lane < 16) ? 1.0f : 0.0f;
  // B (4x16): lanes 0-15 hold col N=lane: K0=1, K1=area_j[N]; lanes 16-31: K2=K3=0
  v2f B; B[0] = (lane < 16) ? 1.0f : 0.0f; B[1] = (lane < 16) ? aj : 0.0f;
  v8f C = {};
  return __builtin_amdgcn_wmma_f32_16x16x4_f32(false, A, false, B, (short)0, C,
                                               false, false);
#else
  v8f S;
  #pragma unroll
  for (int k = 0; k < 8; ++k) S[k] = ai + aj;  // unused fallback
  return S;
#endif
}

__device__ __forceinline__ unsigned wave_ballot(bool p) {
#if __has_builtin(__builtin_amdgcn_ballot_w32)
  return __builtin_amdgcn_ballot_w32(p);
#else
  return (unsigned)__ballot(p);
#endif
}

// IoU -> suppression bitmask. One wave32 per 16x32 (i,j) tile: two 16x16
// sub-tiles in the WMMA f32 16x16 C/D striping (VGPR k <-> rows k/k+8,
// lane <-> column). The 32-lane ballot of the predicate IS the packed mask:
// bits 0-15 = row k's 16 cols, bits 16-31 = row (k+8)'s 16 cols.
__global__ void k_mask(const float4* __restrict__ bsort, const float* __restrict__ area,
                       unsigned* __restrict__ mask) {
  const int lane = threadIdx.x & 31;
  const int wave = threadIdx.x >> 5;
  const int tile = blockIdx.x * 8 + wave;
  const int i_tile = tile >> 8;            // 256 32-col words per tile row
  const int jw     = tile & 255;           // which uint32 mask word (32 cols)
  const int i0 = i_tile << 4;
  if (jw < (i0 >> 5)) return;              // wave-uniform: EXEC stays all-ones;
                                           // scan only reads words w >= i>>5
  const int jA = jw << 5;                  // cols jA..jA+15
  const int jB = jA + 16;                  // cols jB..jB+15

  const int n    = lane & 15;              // column within sub-tile (dup across halves)
  const int half = lane >> 4;              // 0: rows 0..7, 1: rows 8..15

  __builtin_prefetch(bsort + i0, 0, 3);    // near-scope prefetch of i-row boxes

  const float4 bjA = bsort[jA + n];
  const float4 bjB = bsort[jB + n];
  const float  ai  = area[i0 + n];
  const float  ajA = area[jA + n];
  const float  ajB = area[jB + n];

  const v8f SA = area_outer_sum(lane, ai, ajA);
  const v8f SB = area_outer_sum(lane, ai, ajB);

  #pragma unroll
  for (int k = 0; k < 8; ++k) {
    const int gi = i0 + k + half * 8;      // global row index (sorted order)
    const float4 bi = bsort[gi];

    float lxA = fmaxf(bi.x, bjA.x), lyA = fmaxf(bi.y, bjA.y);
    float rxA = fminf(bi.z, bjA.z), ryA = fminf(bi.w, bjA.w);
    float iA  = fmaxf(rxA - lxA, 0.0f) * fmaxf(ryA - lyA, 0.0f);
    bool pA = (iA > IOU_THR * (SA[k] - iA)) && (jA + n > gi);

    float lxB = fmaxf(bi.x, bjB.x), lyB = fmaxf(bi.y, bjB.y);
    float rxB = fminf(bi.z, bjB.z), ryB = fminf(bi.w, bjB.w);
    float iB  = fmaxf(rxB - lxB, 0.0f) * fmaxf(ryB - lyB, 0.0f);
    bool pB = (iB > IOU_THR * (SB[k] - iB)) && (jB + n > gi);

    const unsigned bA = wave_ballot(pA);   // [15:0]=row k, [31:16]=row k+8 (cols jA+)
    const unsigned bB = wave_ballot(pB);   // same for cols jB+
    // row k word: cols jA.. in [15:0], cols jB.. in [31:16]; row k+8 analogous.
    if (lane < 2) {
      unsigned v = lane ? ((bA >> 16) | (bB & 0xffff0000u))
                        : ((bA & 0xffffu) | (bB << 16));
      mask[(size_t)(i0 + k + 8 * lane) * 256 + jw] = v;
    }
  }
}

// sequential greedy scan: removed bitset in LDS; for each un-removed i, OR its
// mask row (words w >= i/32; bits j<=i are zero by construction) into removed.
__global__ void k_scan(const unsigned* __restrict__ mask, int* __restrict__ keep) {
  __shared__ unsigned removed[256];        // 8192 bits
  const int tid = threadIdx.x;
  removed[tid] = 0u;
  __syncthreads();
  for (int i = 0; i < NN; ++i) {
    unsigned bit = (removed[i >> 5] >> (i & 31)) & 1u;
    if (!bit) {
      const unsigned* row = mask + (size_t)i * 256;
      if (tid >= (i >> 5)) removed[tid] |= row[tid];
      // race on word i>>5 is benign: this OR never touches bit i (j>i only)
    }
    __syncthreads();
  }
  for (int i = tid; i < NN; i += 256)
    keep[i] = ((removed[i >> 5] >> (i & 31)) & 1u) ? 0 : 1;
}

// out = X[order] * keep[:,None]
__global__ void k_out(const float* __restrict__ X, const int* __restrict__ order,
                      const int* __restrict__ keep, float* __restrict__ out) {
  int idx = blockIdx.x * blockDim.x + threadIdx.x;
  if (idx >= NN * ROWLEN) return;
  int i = idx / ROWLEN, c = idx - i * ROWLEN;
  out[idx] = keep[i] ? X[(size_t)order[i] * ROWLEN + c] : 0.0f;
}

// ---------------------------------------------------------------------------
extern "C" void kernel_launch(void* const* d_in, const int* in_sizes, int n_in,
                              void* d_out, int out_size, void* d_ws, size_t ws_size,
                              hipStream_t stream) {
  (void)in_sizes; (void)n_in; (void)out_size; (void)ws_size;
  const float* X = (const float*)d_in[0];
  char* ws = (char*)d_ws;
  float*    maxc   = (float*)(ws + OFF_MAXC);
  float*    scores = (float*)(ws + OFF_SCORES);
  int*      cat    = (int*)(ws + OFF_CAT);
  int*      order  = (int*)(ws + OFF_ORDER);
  float*    area   = (float*)(ws + OFF_AREA);
  float4*   bsort  = (float4*)(ws + OFF_BSORT);
  int*      keep   = (int*)(ws + OFF_KEEP);
  unsigned* mask   = (unsigned*)(ws + OFF_MASK);
  float*    out    = (float*)d_out;

  k_init  <<<1, 1, 0, stream>>>((int*)maxc);
  k_scores<<<NN / 256, 256, 0, stream>>>(X, scores, cat, (int*)maxc);
  k_sort  <<<1, 1024, 64 * 1024, stream>>>(scores, order);
  k_gather<<<NN / 256, 256, 0, stream>>>(X, order, cat, maxc, bsort, area);
  const int tiles = (NN / 16) * (NN / 32);           // 131072 16x32 tiles
  k_mask  <<<tiles / 8, 256, 0, stream>>>(bsort, area, mask);
  k_scan  <<<1, 256, 0, stream>>>(mask, keep);
  k_out   <<<(NN * ROWLEN + 255) / 256, 256, 0, stream>>>(X, order, keep, out);
}